// DensityCalculator_52166672777376
// MI455X (gfx1250) — compile-verified
//
#include <hip/hip_runtime.h>
#include <hip/hip_bf16.h>

// density[g] = sum_a mask_a * sum_n aw[e_a,n] * exp(bw[e_a,n] * ||g - X_a||^2)
//
// MI455X (gfx1250, wave32) strategy:
//  - r^2 for 16x16 (grid x atom) tiles via V_WMMA_F32_16X16X4_F32:
//      r2 = |g'|^2 (C) + [g'x g'y g'z 1] (A) x [-2x'; -2y'; -2z'; |x'|^2] (B)
//    coordinates re-centered per wave for numerical stability.
//  - Each wave owns 32 grid rows (two M-tiles sharing one B per atom tile):
//    2 WMMAs + 96 v_exp_f32 per ~17 gathered loads per lane.
//  - exp(b*r2) as one v_exp_f32 via exp2((b*log2e)*r2).
//  - Branchless masking; ds_swizzle xor-butterfly for the final 16-lane
//    atom-column reduction.

typedef __attribute__((ext_vector_type(2))) float v2f;
typedef __attribute__((ext_vector_type(8))) float v8f;

#define LOG2E 1.4426950408889634f
#define EXCLUDED_ELEM 5
#define N_GAUSS 6

// lane ^ MASK within the wave: group-of-32 swizzle, and=0x1f, or=0, xor=MASK
template <int MASK>
__device__ __forceinline__ void butterfly_add(float (&a)[16]) {
#pragma unroll
    for (int v = 0; v < 16; ++v) {
        int s = __builtin_amdgcn_ds_swizzle(__float_as_int(a[v]),
                                            0x1f | (MASK << 10));
        a[v] += __int_as_float(s);
    }
}

__global__ __launch_bounds__(256) void density_wmma_kernel(
    const float* __restrict__ grid,   // [Ng,3]
    const float* __restrict__ X,      // [Na,3]
    const float* __restrict__ aw,     // [6,6]  aw_table
    const float* __restrict__ bw,     // [6,6]  bw_table
    const int*   __restrict__ elem,   // [Na]
    const int*   __restrict__ cexp,   // [Na]
    float*       __restrict__ out,    // [Ng]
    int nAtoms)
{
    const int lane  = threadIdx.x & 31;
    const int wid   = (blockIdx.x * blockDim.x + threadIdx.x) >> 5;
    const int half  = lane >> 4;        // half-wave (K/M split in WMMA layouts)
    const int M     = lane & 15;        // row within M-tile == atom column N
    const int wbase = wid << 5;         // first of 32 grid rows for this wave

    // Re-centering origin: middle of the wave's 32-row span (uniform -> SGPR).
    const float g0x = grid[(wbase + 16) * 3 + 0];
    const float g0y = grid[(wbase + 16) * 3 + 1];
    const float g0z = grid[(wbase + 16) * 3 + 2];

    // Two M-tiles of grid points.
    const int r0 = wbase + M;
    const int r1 = wbase + 16 + M;
    const float g0xM = grid[r0 * 3 + 0] - g0x;
    const float g0yM = grid[r0 * 3 + 1] - g0y;
    const float g0zM = grid[r0 * 3 + 2] - g0z;
    const float g1xM = grid[r1 * 3 + 0] - g0x;
    const float g1yM = grid[r1 * 3 + 1] - g0y;
    const float g1zM = grid[r1 * 3 + 2] - g0z;
    const float gsq0 = g0xM * g0xM + g0yM * g0yM + g0zM * g0zM;
    const float gsq1 = g1xM * g1xM + g1yM * g1yM + g1zM * g1zM;

    // A matrices (16x4 f32): lane holds K = 2*half, 2*half+1 of row M.
    v2f amat0, amat1;
    amat0.x = half ? g0zM : g0xM;
    amat0.y = half ? 1.0f : g0yM;
    amat1.x = half ? g1zM : g1xM;
    amat1.y = half ? 1.0f : g1yM;

    // C matrices: C[m][n] = |g'_m|^2; lane's VGPR v holds m = v + 8*half.
    v8f cmat0, cmat1;
#pragma unroll
    for (int v = 0; v < 8; ++v) {
        cmat0[v] = __shfl(gsq0, 8 * half + v, 32);
        cmat1[v] = __shfl(gsq1, 8 * half + v, 32);
    }

    float acc[16];
#pragma unroll
    for (int v = 0; v < 16; ++v) acc[v] = 0.0f;

    const int nTiles = nAtoms >> 4;
    for (int t = 0; t < nTiles; ++t) {
        const int atom = (t << 4) + M;    // this lane's atom column N = M

        // Eager, unconditional loads (branchless mask below).
        const float xr0 = X[atom * 3 + 0];
        const float xr1 = X[atom * 3 + 1];
        const float xr2 = X[atom * 3 + 2];
        const int   e   = elem[atom];
        const int   ce  = cexp[atom];

        const float xx = xr0 - g0x;
        const float xy = xr1 - g0y;
        const float xz = xr2 - g0z;
        const float xsq = xx * xx + xy * xy + xz * xz;

        // B matrix (4x16 f32): column N = [-2x', -2y', -2z', |x'|^2];
        // lane holds rows K = 2*half, 2*half+1.
        v2f bmat;
        bmat.x = half ? (-2.0f * xz) : (-2.0f * xx);
        bmat.y = half ? xsq          : (-2.0f * xy);

        // Two 16x16 r^2 tiles sharing B.
        v8f r2a = __builtin_amdgcn_wmma_f32_16x16x4_f32(
            false, amat0, false, bmat, (short)0, cmat0, false, false);
        v8f r2b = __builtin_amdgcn_wmma_f32_16x16x4_f32(
            false, amat1, false, bmat, (short)0, cmat1, false, false);

        // Branchless mask: integer & of both predicates -> v_cndmask, no branch.
        const int   ok = (int)(e != EXCLUDED_ELEM) & (int)(ce == 1);
        const float mk = (float)ok;

        float awn[N_GAUSS], b2n[N_GAUSS];
#pragma unroll
        for (int n = 0; n < N_GAUSS; ++n) {
            awn[n] = aw[e * N_GAUSS + n] * mk;
            b2n[n] = bw[e * N_GAUSS + n] * LOG2E;   // exp(b*r2) = exp2(b2*r2)
        }

        // 96 v_exp_f32 + 96 v_fma per lane per tile: the dominant cost.
#pragma unroll
        for (int v = 0; v < 8; ++v) {
            const float ra = r2a[v];
            const float rb = r2b[v];
            float sa = 0.0f, sb = 0.0f;
#pragma unroll
            for (int n = 0; n < N_GAUSS; ++n) {
                sa += awn[n] * __builtin_amdgcn_exp2f(b2n[n] * ra);
                sb += awn[n] * __builtin_amdgcn_exp2f(b2n[n] * rb);
            }
            acc[v]     += sa;
            acc[8 + v] += sb;
        }
    }

    // Sum the 16 atom columns: xor masks 1,2,4,8 stay within each 16-lane half.
    butterfly_add<1>(acc);
    butterfly_add<2>(acc);
    butterfly_add<4>(acc);
    butterfly_add<8>(acc);

    // Lane 0 holds rows {0..7, 16..23}; lane 16 holds rows {8..15, 24..31}.
    if (M == 0) {
#pragma unroll
        for (int v = 0; v < 8; ++v) {
            out[wbase + 8 * half + v]      = acc[v];
            out[wbase + 16 + 8 * half + v] = acc[8 + v];
        }
    }
}

extern "C" void kernel_launch(void* const* d_in, const int* in_sizes, int n_in,
                              void* d_out, int out_size, void* d_ws, size_t ws_size,
                              hipStream_t stream) {
    const float* grid = (const float*)d_in[0];   // [Ng,3]
    const float* X    = (const float*)d_in[1];   // [Na,3]
    const float* aw   = (const float*)d_in[2];   // [6,6]
    const float* bw   = (const float*)d_in[3];   // [6,6]
    const int*   elem = (const int*)d_in[4];     // [Na]
    const int*   cexp = (const int*)d_in[5];     // [Na]
    float* out = (float*)d_out;

    const int nGrid  = in_sizes[0] / 3;          // 262144
    const int nAtoms = in_sizes[1] / 3;          // 2048

    // 32 grid rows per wave, 8 waves (256 threads) per block -> 256 rows/block.
    const int rowsPerBlock = 256;
    const int nBlocks = (nGrid + rowsPerBlock - 1) / rowsPerBlock;

    density_wmma_kernel<<<nBlocks, 256, 0, stream>>>(
        grid, X, aw, bw, elem, cexp, out, nAtoms);
}